// NoisyTopKRouter_11639361372756
// MI455X (gfx1250) — compile-verified
//
#include <hip/hip_runtime.h>
#include <hip/hip_bf16.h>
#include <math.h>

// ---------------------------------------------------------------------------
// NoisyTopKRouter on gfx1250 (MI455X).
//   clean  = x @ w_gate    [N,8]
//   raw    = x @ w_noise   [N,8]
//   noisy  = clean + (softplus(raw)+0.2)*eps
//   gates  = scatter(softmax(top2(noisy)))        -> d_out[0 .. N*8)
//   aux    = 8 * sum_e mean(softmax(clean))_e * mean(gate_e>0)  -> d_out[N*8]
//
// GEMM via V_WMMA_F32_16X16X4_F32:
//   A (16x4)  = packed W^T tile : M = expert (0..7 gate, 8..15 noise), K = dim
//   B (4x16)  = x^T tile        : K = dim, N = token
//   D (16x16) : lane L (0..15)  holds clean logits e=0..7 of token L in acc[0..7]
//               lane L (16..31) holds raw-noise logits of token L-16
//
// Data movement: double-buffered KT=32 tiles, streamed with
// GLOBAL_LOAD_ASYNC_TO_LDS_B128 (ASYNCcnt) when available, so tile t+1 is in
// flight behind tile t's WMMAs.  Fallback: b128 global->VGPR->ds_store_b128.
// ---------------------------------------------------------------------------

typedef float v2f __attribute__((ext_vector_type(2)));
typedef float v8f __attribute__((ext_vector_type(8)));
typedef int   v4i_vs __attribute__((vector_size(16)));   // matches builtin param

#define EMBED_DIM    1024
#define NUM_EXPERTS  8
#define KT           32                 // K-tile
#define NTILES       (EMBED_DIM / KT)   // 32
#define XT_STRIDE    44                 // 32+12 pad: 176B rows (16B aligned), all-64-bank b64 reads
#define TOK_PER_WAVE 16
#define WAVES_PER_BLOCK 8
#define TOK_PER_BLOCK (TOK_PER_WAVE * WAVES_PER_BLOCK)   // 128

#if __has_builtin(__builtin_amdgcn_global_load_async_to_lds_b128) && \
    __has_builtin(__builtin_amdgcn_s_wait_asynccnt)
#define USE_ASYNC_LDS 1
#else
#define USE_ASYNC_LDS 0
#endif

#if USE_ASYNC_LDS
// gfx1250: generic pointer to LDS carries the LDS byte offset in its low 32
// bits (ISA 10.2); global generic == as(1) value.
#define TO_LDS(p) ((__attribute__((address_space(3))) v4i_vs*)(unsigned int)(unsigned long long)(p))
#define TO_GLB(p) ((__attribute__((address_space(1))) v4i_vs*)(unsigned long long)(p))
#define ASYNC_B128(g, l) \
    __builtin_amdgcn_global_load_async_to_lds_b128(TO_GLB(g), TO_LDS(l), 0, 0)
#define WAIT_ASYNC(n) __builtin_amdgcn_s_wait_asynccnt(n)
#else
#define WAIT_ASYNC(n)
#endif

__device__ __forceinline__ float softplus_f(float x) {
    // stable log(1+exp(x))
    return fmaxf(x, 0.0f) + log1pf(expf(-fabsf(x)));
}

__global__ __launch_bounds__(256)
void router_main_kernel(const float* __restrict__ x,
                        const float* __restrict__ eps,
                        const float* __restrict__ w_gate,
                        const float* __restrict__ w_noise,
                        float* __restrict__ gates,       // d_out, [N,8]
                        float* __restrict__ ws_partials) // [gridDim.x * 16]
{
    __shared__ float Wt[2][KT * 16];                                   // 4 KB
    __shared__ float Xt[2][WAVES_PER_BLOCK][TOK_PER_WAVE * XT_STRIDE]; // 44 KB
    __shared__ float Epi[TOK_PER_BLOCK * 16];                          // 8 KB

    const int tid  = threadIdx.x;
    const int lane = tid & 31;
    const int wave = tid >> 5;
    const int tokBase = blockIdx.x * TOK_PER_BLOCK + wave * TOK_PER_WAVE;

    const int n    = lane & 15;        // expert (A) / token (B) / column (D)
    const int koff = (lane >> 4) * 2;  // K sub-offset from WMMA lane layout

    // x-tile loader lane mapping: 4 b128 per lane covers 16 rows x 32 floats
    const int xlr = lane >> 3;         // row sub-index 0..3
    const int xlc = (lane & 7) * 4;    // float col offset within tile

    // W-tile loader: slot tid&127 (issued redundantly by upper half so every
    // wave has a uniform ASYNCcnt of 5 per tile; duplicate writes are benign)
    const int wslot = tid & 127;
    const int wd = wslot >> 2;         // dim row 0..31
    const int wq = wslot & 3;          // vec4 quarter of the 16-wide row

    const float* xrow = x + (size_t)tokBase * EMBED_DIM;
    v8f acc = {};

    // ---- tile issue helpers -------------------------------------------------
    auto issue_x = [&](int kt, int buf) {
        float* Xb = &Xt[buf][wave][0];
        #pragma unroll
        for (int j = 0; j < 4; ++j) {
            int row  = j * 4 + xlr;
            const float* g = xrow + (size_t)row * EMBED_DIM + kt + xlc;
            float* l = Xb + row * XT_STRIDE + xlc;
#if USE_ASYNC_LDS
            ASYNC_B128(g, l);
#else
            float4 v = *(const float4*)g;
            *(float4*)l = v;
#endif
        }
    };
    auto issue_w = [&](int kt, int buf) {
        const float* g = (wq < 2)
            ? (w_gate  + (size_t)(kt + wd) * NUM_EXPERTS + wq * 4)
            : (w_noise + (size_t)(kt + wd) * NUM_EXPERTS + (wq - 2) * 4);
        float* l = &Wt[buf][wd * 16 + wq * 4];
#if USE_ASYNC_LDS
        ASYNC_B128(g, l);
#else
        float4 v = *(const float4*)g;
        *(float4*)l = v;
#endif
    };

    // ---- software pipeline: tile t computes while tile t+1 streams in ------
    issue_x(0, 0);
    __syncthreads();           // W buffer 0 free for everyone
    issue_w(0, 0);

    for (int t = 0; t < NTILES; ++t) {
        const int buf = t & 1;
        const bool more = (t + 1) < NTILES;

        if (more) issue_x((t + 1) * KT, buf ^ 1);   // private buffer: safe now
        __syncthreads();                            // all waves done reading W[buf^1]
        if (more) {
            issue_w((t + 1) * KT, buf ^ 1);
            WAIT_ASYNC(5);     // 5 just issued for t+1 -> all of tile t resident
        } else {
            WAIT_ASYNC(0);
        }
        __syncthreads();                            // publish tile t to all waves

        const float* Wb = &Wt[buf][0];
        const float* Xw = &Xt[buf][wave][0];
        #pragma unroll
        for (int kk = 0; kk < KT; kk += 4) {
            v2f a, b;
            a.x = Wb[(kk + koff    ) * 16 + n];
            a.y = Wb[(kk + koff + 1) * 16 + n];
            b   = *(const v2f*)(&Xw[n * XT_STRIDE + kk + koff]);
            acc = __builtin_amdgcn_wmma_f32_16x16x4_f32(
                      false, a, false, b, (short)0, acc, false, false);
        }
    }

    // ---- pull raw-noise logits (lanes 16..31) down to lanes 0..15
    float nraw[8];
    #pragma unroll
    for (int e = 0; e < 8; ++e) {
        nraw[e] = __int_as_float(
            __builtin_amdgcn_ds_bpermute(((lane + 16) & 31) << 2,
                                         __float_as_int(acc[e])));
    }

    if (lane < 16) {
        const int tok = tokBase + lane;

        float4 e0 = ((const float4*)eps)[(size_t)tok * 2 + 0];
        float4 e1 = ((const float4*)eps)[(size_t)tok * 2 + 1];
        float ev[8] = {e0.x, e0.y, e0.z, e0.w, e1.x, e1.y, e1.z, e1.w};

        float clean[8], nz[8];
        #pragma unroll
        for (int e = 0; e < 8; ++e) {
            clean[e] = acc[e];
            float stdv = softplus_f(nraw[e]) + 0.2f;   // NOISE_EPS
            nz[e] = clean[e] + stdv * ev[e];
        }

        // top-2 (strict '>' => lowest index wins ties, matching jax top_k)
        int i0 = 0; float v0 = nz[0];
        #pragma unroll
        for (int e = 1; e < 8; ++e) if (nz[e] > v0) { v0 = nz[e]; i0 = e; }
        int i1 = (i0 == 0) ? 1 : 0; float v1 = nz[i1];
        #pragma unroll
        for (int e = 0; e < 8; ++e)
            if (e != i0 && e != i1 && nz[e] > v1) { v1 = nz[e]; i1 = e; }

        // softmax over the two selected logits (v0 >= v1 -> stable)
        float t1 = expf(v1 - v0);
        float g1 = t1 / (1.0f + t1);
        float g0 = 1.0f / (1.0f + t1);

        float g[8];
        #pragma unroll
        for (int e = 0; e < 8; ++e) g[e] = 0.0f;
        g[i0] = g0; g[i1] = g1;

        float4 o0 = make_float4(g[0], g[1], g[2], g[3]);
        float4 o1 = make_float4(g[4], g[5], g[6], g[7]);
        ((float4*)gates)[(size_t)tok * 2 + 0] = o0;
        ((float4*)gates)[(size_t)tok * 2 + 1] = o1;

        // softmax over clean logits (for P) + topk indicator (for f)
        float m = clean[0];
        #pragma unroll
        for (int e = 1; e < 8; ++e) m = fmaxf(m, clean[e]);
        float pe[8], s = 0.0f;
        #pragma unroll
        for (int e = 0; e < 8; ++e) { pe[e] = expf(clean[e] - m); s += pe[e]; }
        float inv = 1.0f / s;

        int r = wave * TOK_PER_WAVE + lane;
        #pragma unroll
        for (int e = 0; e < 8; ++e) Epi[r * 16 + e] = pe[e] * inv;
        #pragma unroll
        for (int e = 0; e < 8; ++e)
            Epi[r * 16 + 8 + e] =
                (e == i0 || (e == i1 && g1 > 0.0f)) ? 1.0f : 0.0f;
    }

    __syncthreads();

    // deterministic per-block partial sums: 16 columns, serial over 128 tokens
    if (tid < 16) {
        float s = 0.0f;
        #pragma unroll 4
        for (int r = 0; r < TOK_PER_BLOCK; ++r) s += Epi[r * 16 + tid];
        ws_partials[(size_t)blockIdx.x * 16 + tid] = s;
    }
}

__global__ void router_finalize_kernel(const float* __restrict__ ws_partials,
                                       float* __restrict__ out,
                                       int nblocks, int ntokens)
{
    __shared__ float tot[16];
    int t = threadIdx.x;
    if (t < 16) {
        float s = 0.0f;
        for (int b = 0; b < nblocks; ++b) s += ws_partials[(size_t)b * 16 + t];
        tot[t] = s;
    }
    __syncthreads();
    if (t == 0) {
        float invN = 1.0f / (float)ntokens;
        float aux = 0.0f;
        #pragma unroll
        for (int e = 0; e < 8; ++e)
            aux += (tot[e] * invN) * (tot[8 + e] * invN);
        out[(size_t)ntokens * NUM_EXPERTS] = (float)NUM_EXPERTS * aux;
    }
}

extern "C" void kernel_launch(void* const* d_in, const int* in_sizes, int n_in,
                              void* d_out, int out_size, void* d_ws, size_t ws_size,
                              hipStream_t stream) {
    const float* x       = (const float*)d_in[0];
    const float* eps     = (const float*)d_in[1];
    const float* w_gate  = (const float*)d_in[2];
    const float* w_noise = (const float*)d_in[3];
    float* out = (float*)d_out;
    float* ws  = (float*)d_ws;

    const int ntokens = in_sizes[0] / EMBED_DIM;     // 65536
    const int nblocks = ntokens / TOK_PER_BLOCK;     // 512

    router_main_kernel<<<dim3(nblocks), dim3(256), 0, stream>>>(
        x, eps, w_gate, w_noise, out, ws);
    router_finalize_kernel<<<dim3(1), dim3(32), 0, stream>>>(
        ws, out, nblocks, ntokens);
}